// MPNN_17961553232388
// MI455X (gfx1250) — compile-verified
//
#include <hip/hip_runtime.h>
#include <math.h>

// ============================================================================
// MPNN forward for MI455X (gfx1250, wave32, WMMA).
//
// Key restructure: never materialize the [B,N,N,50,50] filter tensor (655MB).
//   filters = H @ W_last  (H = 4th hidden layer of filter MLP, [65536,50])
//   msgs[b,i,d] = sum_{k,e} W_last[k, d*50+e] * G[b,i,k,e]
//   G[b,i]      = H[b,i,:,:]^T @ (hs*m)[b]          (per (b,i): [50,50])
// plus a per-batch bias term  sum_e bias[d,e] * (sum_j x[b,j,e]).
// Everything stays L2-resident (192MB L2); all GEMMs run on
// V_WMMA_F32_16X16X4_F32 (native fp32 matrix op -> no precision loss).
//
// Input order assumption (setup_inputs() dict insertion order, depth-first):
//  0 atoms[64,32,25]  1 dist[64,32,32]  2 mask[64,32]
//  3..7 filt_W[0..4]  8..12 filt_b[0..4]
//  13 Wz 14 Uz 15 bz 16 Wr 17 Ur 18 br 19 Wh 20 Uh 21 bh
//  22 lstm_W[225,300] 23 lstm_b[300] 24 W_out1[150,200] 25 b_out1
//  26 W_out2[200,12]  27 b_out2
// ============================================================================

typedef __attribute__((ext_vector_type(2))) float v2f;
typedef __attribute__((ext_vector_type(8))) float v8f;

#define NB    64      // batch
#define NA    32      // atoms
#define DH    50      // hidden dim
#define DP    64      // padded hidden dim
#define NROW  65536   // B*N*N rows for filter MLP
#define BN    2048    // B*N rows

__device__ __forceinline__ float activ(float v, int act) {
  if (act == 1) return v > 0.f ? v : 0.01f * v;                 // leaky_relu
  if (act == 2) return 1.f / (1.f + expf(-v));                  // sigmoid
  if (act == 3) return tanhf(v);                                // tanh
  return v;
}

// ---------------------------------------------------------------------------
// Generic fp32 WMMA GEMM:  Out = act( A[M,K] @ B[K,N] (+Cin) (+bias_row) )
// Row-major, N % 16 == 0, K % 4 == 0, M % 16 == 0. One wave per 16x16 tile.
// Fragment layouts per CDNA5 ISA 7.12.2:
//   A 16x4 : lane l (0..15)->M=l, (16..31)->M=l-16; v[0]=K(2*half), v[1]=+1
//   B 4x16 : v[0]: rows K=0|K=2 ; v[1]: rows K=1|K=3 (lanes 0-15 | 16-31)
//   C 16x16: vgpr r: lanes0-15 M=r, lanes16-31 M=r+8
// ---------------------------------------------------------------------------
__global__ void gemm_f32_wmma(const float* __restrict__ A, int lda,
                              const float* __restrict__ Bm, int ldb,
                              const float* __restrict__ Cin,
                              const float* __restrict__ bias,
                              float* __restrict__ Out, int ldc,
                              int M, int N, int K, int act) {
  const int lane = threadIdx.x & 31;
  const int half = lane >> 4;
  const int l16  = lane & 15;
  const int ntN  = N >> 4;
  const int tiles = (M >> 4) * ntN;
  const int wave = blockIdx.x * (blockDim.x >> 5) + (threadIdx.x >> 5);
  if (wave >= tiles) return;                 // wave-uniform: EXEC stays full
  const int mt = wave / ntN, nt = wave % ntN;
  const int row0 = mt << 4, col0 = nt << 4;

  v8f acc = {0.f, 0.f, 0.f, 0.f, 0.f, 0.f, 0.f, 0.f};
  const float* ap = A  + (size_t)(row0 + l16) * lda + 2 * half;
  const float* bp = Bm + (size_t)(2 * half) * ldb + col0 + l16;
  for (int k0 = 0; k0 < K; k0 += 4) {
    v2f av, bv;
    av[0] = ap[k0];
    av[1] = ap[k0 + 1];
    bv[0] = bp[(size_t)k0 * ldb];
    bv[1] = bp[(size_t)(k0 + 1) * ldb];
    acc = __builtin_amdgcn_wmma_f32_16x16x4_f32(false, av, false, bv,
                                                (short)0, acc, false, false);
  }
#pragma unroll
  for (int r = 0; r < 8; ++r) {
    const int row = row0 + r + 8 * half;
    const int col = col0 + l16;
    float v = acc[r];
    if (Cin)  v += Cin[(size_t)row * ldc + col];
    if (bias) v += bias[col];
    Out[(size_t)row * ldc + col] = activ(v, act);
  }
}

// ---------------------------------------------------------------------------
// G[b,i] = H[b,i,:,:]^T @ x[b]   -> per (b,i) a 64x64 padded tile.
// M = k index (of H cols), N = e index (of x cols), K = j = 32.
// One wave per 16x16 tile, 16 tiles per (b,i), 32768 waves total.
// ---------------------------------------------------------------------------
__global__ void g_mm_kernel(const float* __restrict__ H,
                            const float* __restrict__ x,
                            float* __restrict__ G) {
  const int lane = threadIdx.x & 31;
  const int half = lane >> 4;
  const int l16  = lane & 15;
  const int w = blockIdx.x * (blockDim.x >> 5) + (threadIdx.x >> 5);
  if (w >= BN * 16) return;
  const int bi = w >> 4;
  const int t  = w & 15, tk = t >> 2, te = t & 3;
  const int b  = bi >> 5;

  // A[m=k][kj=j] = H[(bi*32 + j)*64 + k]
  const float* hb = H + (size_t)(bi * NA + 2 * half) * DP + tk * 16 + l16;
  // B[j][e] = x[(b*32 + j)*64 + e]
  const float* xb = x + (size_t)(b * NA + 2 * half) * DP + te * 16 + l16;

  v8f acc = {0.f, 0.f, 0.f, 0.f, 0.f, 0.f, 0.f, 0.f};
  for (int j0 = 0; j0 < NA; j0 += 4) {
    v2f av, bv;
    const float* hp = hb + (size_t)j0 * DP;
    av[0] = hp[0];  av[1] = hp[DP];
    const float* xp = xb + (size_t)j0 * DP;
    bv[0] = xp[0];  bv[1] = xp[DP];
    acc = __builtin_amdgcn_wmma_f32_16x16x4_f32(false, av, false, bv,
                                                (short)0, acc, false, false);
  }
  float* gp = G + (size_t)bi * (DP * DP);
#pragma unroll
  for (int r = 0; r < 8; ++r)
    gp[(size_t)(tk * 16 + r + 8 * half) * DP + te * 16 + l16] = acc[r];
}

// ---------------------------------------------------------------------------
// Small elementwise / repack kernels
// ---------------------------------------------------------------------------
__global__ void pad2d(const float* __restrict__ src, int sr, int sc,
                      float* __restrict__ dst, int dr, int dc) {
  const int i = blockIdx.x * blockDim.x + threadIdx.x;
  if (i >= dr * dc) return;
  const int r = i / dc, c = i % dc;
  dst[i] = (r < sr && c < sc) ? src[r * sc + c] : 0.f;
}

// W2p[(k*64+e)*64 + d] = W_last[k*2500 + d*50 + e]   (zero-padded)
__global__ void pack_w2(const float* __restrict__ Wlast, float* __restrict__ W2p) {
  const int i = blockIdx.x * blockDim.x + threadIdx.x;
  if (i >= 4096 * 64) return;
  const int d = i & 63, ke = i >> 6, e = ke & 63, k = ke >> 6;
  W2p[i] = (k < DH && e < DH && d < DH) ? Wlast[k * 2500 + d * DH + e] : 0.f;
}

// filter MLP layer0: out[r,c] = leaky(dist[r]*W0[c] + b0[c]) for c<50 else 0
__global__ void filt_layer0(const float* __restrict__ dist,
                            const float* __restrict__ W0,
                            const float* __restrict__ b0,
                            float* __restrict__ out) {
  const int i = blockIdx.x * blockDim.x + threadIdx.x;
  if (i >= NROW * DP) return;
  const int c = i & 63, r = i >> 6;
  float v = 0.f;
  if (c < DH) { v = dist[r] * W0[c] + b0[c]; v = v > 0.f ? v : 0.01f * v; }
  out[i] = v;
}

__global__ void hs_init(const float* __restrict__ atoms, float* __restrict__ hs) {
  const int i = blockIdx.x * blockDim.x + threadIdx.x;
  if (i >= BN * DP) return;
  const int c = i & 63, r = i >> 6;
  hs[i] = (c < 25) ? atoms[r * 25 + c] : 0.f;
}

__global__ void ew_x(const float* __restrict__ hs, const float* __restrict__ mask,
                     float* __restrict__ x) {
  const int i = blockIdx.x * blockDim.x + threadIdx.x;
  if (i >= BN * DP) return;
  x[i] = hs[i] * mask[i >> 6];
}

// msgsBias[b,d] = sum_e bfilt[d*50+e] * (sum_j x[b,j,e]);  block=64 thr per b
__global__ void msgs_bias(const float* __restrict__ x,
                          const float* __restrict__ bfilt,
                          float* __restrict__ mb) {
  __shared__ float Xs[DP];
  const int b = blockIdx.x, t = threadIdx.x;
  float s = 0.f;
  for (int j = 0; j < NA; ++j) s += x[(size_t)(b * NA + j) * DP + t];
  Xs[t] = s;
  __syncthreads();
  float v = 0.f;
  if (t < DH)
    for (int e = 0; e < DH; ++e) v += bfilt[t * DH + e] * Xs[e];
  mb[b * DP + t] = v;
}

__global__ void add_bias(float* __restrict__ msgs, const float* __restrict__ mb) {
  const int i = blockIdx.x * blockDim.x + threadIdx.x;
  if (i >= BN * DP) return;
  const int row = i >> 6, b = row >> 5;
  msgs[i] += mb[b * DP + (i & 63)];
}

__global__ void ew_rh(const float* __restrict__ R, const float* __restrict__ hs,
                      float* __restrict__ RH) {
  const int i = blockIdx.x * blockDim.x + threadIdx.x;
  if (i >= BN * DP) return;
  RH[i] = R[i] * hs[i];
}

__global__ void ew_hsupd(const float* __restrict__ Z, const float* __restrict__ HC,
                         const float* __restrict__ mask, float* __restrict__ hs) {
  const int i = blockIdx.x * blockDim.x + threadIdx.x;
  if (i >= BN * DP) return;
  const float z = Z[i];
  hs[i] = ((1.f - z) * hs[i] + z * HC[i]) * mask[i >> 6];
}

// feats[bn, 0..24]=atoms, [25..74]=hs, [75..79]=0   (P=75 padded to 80)
__global__ void feats_build(const float* __restrict__ atoms,
                            const float* __restrict__ hs,
                            float* __restrict__ feats) {
  const int i = blockIdx.x * blockDim.x + threadIdx.x;
  if (i >= BN * 80) return;
  const int r = i / 80, c = i % 80;
  float v = 0.f;
  if (c < 25)      v = atoms[r * 25 + c];
  else if (c < 75) v = hs[(size_t)r * DP + (c - 25)];
  feats[i] = v;
}

__global__ void s2v_init(float* __restrict__ q, float* __restrict__ c,
                         float* __restrict__ qstar) {
  const int i = blockIdx.x * blockDim.x + threadIdx.x;
  if (i < NB * 80)  { q[i] = 0.f; c[i] = 0.f; }
  if (i < NB * 160) qstar[i] = 0.f;
}

// cat[b, 0..149]=q_star, [150..224]=q, [225..239]=0
__global__ void s2v_concat(const float* __restrict__ qstar,
                           const float* __restrict__ q, float* __restrict__ cat) {
  const int i = blockIdx.x * blockDim.x + threadIdx.x;
  if (i >= NB * 240) return;
  const int b = i / 240, cix = i % 240;
  float v = 0.f;
  if (cix < 150)      v = qstar[b * 160 + cix];
  else if (cix < 225) v = q[b * 80 + (cix - 150)];
  cat[i] = v;
}

// gates[b,304]: i=0..74 f=75..149 g=150..224 o=225..299
__global__ void s2v_lstm(const float* __restrict__ gates, float* __restrict__ c,
                         float* __restrict__ q) {
  const int i = blockIdx.x * blockDim.x + threadIdx.x;
  if (i >= NB * 80) return;
  const int b = i / 80, p = i % 80;
  float cn = 0.f, qn = 0.f;
  if (p < 75) {
    const float* g = gates + (size_t)b * 304;
    const float ig = 1.f / (1.f + expf(-g[p]));
    const float fg = 1.f / (1.f + expf(-g[75 + p]));
    const float gg = tanhf(g[150 + p]);
    const float og = 1.f / (1.f + expf(-g[225 + p]));
    cn = fg * c[i] + ig * gg;
    qn = og * tanhf(cn);
  }
  c[i] = cn;
  q[i] = qn;
}

// attention + readout: one block (160 thr) per batch b
__global__ void s2v_attn(const float* __restrict__ feats,
                         const float* __restrict__ q,
                         const float* __restrict__ mask,
                         float* __restrict__ qstar) {
  __shared__ float abuf[NA];
  const int b = blockIdx.x, t = threadIdx.x;
  if (t < NA) {
    float s = 0.f;
    const float* fp = feats + (size_t)(b * NA + t) * 80;
    for (int p = 0; p < 75; ++p) s += fp[p] * q[b * 80 + p];
    abuf[t] = (mask[b * NA + t] > 0.f) ? s : -1e9f;
  }
  __syncthreads();
  if (t == 0) {
    float mx = -1e30f;
    for (int n = 0; n < NA; ++n) mx = fmaxf(mx, abuf[n]);
    float sum = 0.f;
    for (int n = 0; n < NA; ++n) { abuf[n] = expf(abuf[n] - mx); sum += abuf[n]; }
    const float inv = 1.f / sum;
    for (int n = 0; n < NA; ++n) abuf[n] *= inv;
  }
  __syncthreads();
  if (t < 75) {
    qstar[b * 160 + t] = q[b * 80 + t];
  } else if (t < 150) {
    const int p = t - 75;
    float s = 0.f;
    for (int n = 0; n < NA; ++n) s += abuf[n] * feats[(size_t)(b * NA + n) * 80 + p];
    qstar[b * 160 + t] = s;
  } else {
    qstar[b * 160 + t] = 0.f;
  }
}

__global__ void out_copy(const float* __restrict__ outF, float* __restrict__ out) {
  const int i = blockIdx.x * blockDim.x + threadIdx.x;
  if (i >= NB * 12) return;
  out[i] = outF[(i / 12) * 16 + (i % 12)];
}

// ===========================================================================
extern "C" void kernel_launch(void* const* d_in, const int* in_sizes, int n_in,
                              void* d_out, int out_size, void* d_ws, size_t ws_size,
                              hipStream_t stream) {
  const float* atoms = (const float*)d_in[0];
  const float* dist  = (const float*)d_in[1];
  const float* mask  = (const float*)d_in[2];
  const float* fW[5] = {(const float*)d_in[3], (const float*)d_in[4],
                        (const float*)d_in[5], (const float*)d_in[6],
                        (const float*)d_in[7]};
  const float* fb[5] = {(const float*)d_in[8], (const float*)d_in[9],
                        (const float*)d_in[10], (const float*)d_in[11],
                        (const float*)d_in[12]};
  const float* Wz = (const float*)d_in[13]; const float* Uz = (const float*)d_in[14];
  const float* bz = (const float*)d_in[15];
  const float* Wr = (const float*)d_in[16]; const float* Ur = (const float*)d_in[17];
  const float* br = (const float*)d_in[18];
  const float* Wh = (const float*)d_in[19]; const float* Uh = (const float*)d_in[20];
  const float* bh = (const float*)d_in[21];
  const float* Wl = (const float*)d_in[22]; const float* bl = (const float*)d_in[23];
  const float* Wo1 = (const float*)d_in[24]; const float* bo1 = (const float*)d_in[25];
  const float* Wo2 = (const float*)d_in[26]; const float* bo2 = (const float*)d_in[27];

  // ---- workspace carve-up (floats) ----
  float* ws = (float*)d_ws;
  size_t off = 0;
  auto alloc = [&](size_t n) { float* p = ws + off; off += (n + 63) & ~(size_t)63; return p; };
  float* Ha   = alloc((size_t)NROW * DP);
  float* Hb   = alloc((size_t)NROW * DP);
  float* G    = alloc((size_t)BN * DP * DP);
  float* W2p  = alloc(4096 * 64);
  float* fW1p = alloc(52 * 64); float* fW2p = alloc(52 * 64); float* fW3p = alloc(52 * 64);
  float* Wzp = alloc(52 * 64); float* Uzp = alloc(52 * 64);
  float* Wrp = alloc(52 * 64); float* Urp = alloc(52 * 64);
  float* Whp = alloc(52 * 64); float* Uhp = alloc(52 * 64);
  float* b1p = alloc(64); float* b2p = alloc(64); float* b3p = alloc(64);
  float* bzp = alloc(64); float* brp = alloc(64); float* bhp = alloc(64);
  float* Wlp  = alloc(228 * 304); float* blp  = alloc(304);
  float* Wo1p = alloc(152 * 208); float* bo1p = alloc(208);
  float* Wo2p = alloc(200 * 16);  float* bo2p = alloc(16);
  float* hs   = alloc(BN * DP);  float* xb  = alloc(BN * DP);
  float* msgs = alloc(BN * DP);  float* T1  = alloc(BN * DP);
  float* Zb   = alloc(BN * DP);  float* Rb  = alloc(BN * DP);
  float* HCb  = alloc(BN * DP);  float* RHb = alloc(BN * DP);
  float* mB   = alloc(64 * 64);
  float* feats = alloc((size_t)BN * 80);
  float* catb  = alloc(NB * 240); float* gates = alloc(NB * 304);
  float* qb = alloc(NB * 80); float* cb = alloc(NB * 80); float* qstar = alloc(NB * 160);
  float* o1 = alloc(NB * 208); float* oF = alloc(NB * 16);

  auto nb = [](long n, int t) { return (int)((n + t - 1) / t); };
  auto gemm = [&](const float* A, int lda, const float* Bm, int ldb,
                  const float* Cin, const float* bias, float* Out, int ldc,
                  int M, int N, int K, int act) {
    const int tiles = (M / 16) * (N / 16);
    gemm_f32_wmma<<<nb(tiles, 8), 256, 0, stream>>>(A, lda, Bm, ldb, Cin, bias,
                                                    Out, ldc, M, N, K, act);
  };
  auto pad = [&](const float* s, int sr, int sc, float* d, int dr, int dc) {
    pad2d<<<nb((long)dr * dc, 256), 256, 0, stream>>>(s, sr, sc, d, dr, dc);
  };

  // ---- weight repacks (zero-padded, every launch: deterministic) ----
  pad(fW[1], 50, 50, fW1p, 52, 64); pad(fW[2], 50, 50, fW2p, 52, 64);
  pad(fW[3], 50, 50, fW3p, 52, 64);
  pad(Wz, 50, 50, Wzp, 52, 64); pad(Uz, 50, 50, Uzp, 52, 64);
  pad(Wr, 50, 50, Wrp, 52, 64); pad(Ur, 50, 50, Urp, 52, 64);
  pad(Wh, 50, 50, Whp, 52, 64); pad(Uh, 50, 50, Uhp, 52, 64);
  pad(fb[1], 1, 50, b1p, 1, 64); pad(fb[2], 1, 50, b2p, 1, 64);
  pad(fb[3], 1, 50, b3p, 1, 64);
  pad(bz, 1, 50, bzp, 1, 64); pad(br, 1, 50, brp, 1, 64); pad(bh, 1, 50, bhp, 1, 64);
  pad(Wl, 225, 300, Wlp, 228, 304);  pad(bl, 1, 300, blp, 1, 304);
  pad(Wo1, 150, 200, Wo1p, 152, 208); pad(bo1, 1, 200, bo1p, 1, 208);
  pad(Wo2, 200, 12, Wo2p, 200, 16);   pad(bo2, 1, 12, bo2p, 1, 16);
  pack_w2<<<nb(4096 * 64, 256), 256, 0, stream>>>(fW[4], W2p);

  // ---- filter MLP hidden layers (H never hits the 50->2500 layer) ----
  filt_layer0<<<nb((long)NROW * DP, 256), 256, 0, stream>>>(dist, fW[0], fb[0], Ha);
  gemm(Ha, DP, fW1p, 64, nullptr, b1p, Hb, DP, NROW, DP, 52, 1);
  gemm(Hb, DP, fW2p, 64, nullptr, b2p, Ha, DP, NROW, DP, 52, 1);
  gemm(Ha, DP, fW3p, 64, nullptr, b3p, Hb, DP, NROW, DP, 52, 1);  // H = Hb

  // ---- message passing + GRU (3 steps) ----
  hs_init<<<nb(BN * DP, 256), 256, 0, stream>>>(atoms, hs);
  for (int s = 0; s < 3; ++s) {
    ew_x<<<nb(BN * DP, 256), 256, 0, stream>>>(hs, mask, xb);
    g_mm_kernel<<<BN * 16 / 8, 256, 0, stream>>>(Hb, xb, G);
    gemm(G, DP * DP, W2p, 64, nullptr, nullptr, msgs, DP, BN, DP, DP * DP, 0);
    msgs_bias<<<NB, 64, 0, stream>>>(xb, fb[4], mB);
    add_bias<<<nb(BN * DP, 256), 256, 0, stream>>>(msgs, mB);
    // z = sigmoid(msgs@Wz + hs@Uz + bz)
    gemm(msgs, DP, Wzp, 64, nullptr, nullptr, T1, DP, BN, DP, 52, 0);
    gemm(hs, DP, Uzp, 64, T1, bzp, Zb, DP, BN, DP, 52, 2);
    // r = sigmoid(msgs@Wr + hs@Ur + br)
    gemm(msgs, DP, Wrp, 64, nullptr, nullptr, T1, DP, BN, DP, 52, 0);
    gemm(hs, DP, Urp, 64, T1, brp, Rb, DP, BN, DP, 52, 2);
    // hc = tanh(msgs@Wh + (r*hs)@Uh + bh)
    ew_rh<<<nb(BN * DP, 256), 256, 0, stream>>>(Rb, hs, RHb);
    gemm(msgs, DP, Whp, 64, nullptr, nullptr, T1, DP, BN, DP, 52, 0);
    gemm(RHb, DP, Uhp, 64, T1, bhp, HCb, DP, BN, DP, 52, 3);
    ew_hsupd<<<nb(BN * DP, 256), 256, 0, stream>>>(Zb, HCb, mask, hs);
  }

  // ---- Set2Vec readout ----
  feats_build<<<nb((long)BN * 80, 256), 256, 0, stream>>>(atoms, hs, feats);
  s2v_init<<<nb(NB * 160, 256), 256, 0, stream>>>(qb, cb, qstar);
  for (int s = 0; s < 12; ++s) {
    s2v_concat<<<nb(NB * 240, 256), 256, 0, stream>>>(qstar, qb, catb);
    gemm(catb, 240, Wlp, 304, nullptr, blp, gates, 304, NB, 304, 228, 0);
    s2v_lstm<<<nb(NB * 80, 256), 256, 0, stream>>>(gates, cb, qb);
    s2v_attn<<<NB, 160, 0, stream>>>(feats, qb, mask, qstar);
  }

  // ---- output MLP ----
  gemm(qstar, 160, Wo1p, 208, nullptr, bo1p, o1, 208, NB, 208, 152, 1);
  gemm(o1, 208, Wo2p, 16, nullptr, bo2p, oF, 16, NB, 16, 200, 0);
  out_copy<<<nb(NB * 12, 256), 256, 0, stream>>>(oF, (float*)d_out);
}